// Model_32212254720222
// MI455X (gfx1250) — compile-verified
//
#include <hip/hip_runtime.h>
#include <hip/hip_bf16.h>

#define PAGE_SHIFT 4      // page_size = 16
#define PAGE_MASK  15
#define SCAN_THREADS 1024

// ---------------------------------------------------------------------------
// Kernel 1: joint exclusive prefix sum of (ext, new_pages) over B sequences.
// Packs both scans into one u64 (low 32: cumsum(ext), high 32: cumsum(new_pages));
// totals fit comfortably in 32 bits each, so no cross-half carry is possible.
// Single workgroup, Hillis-Steele scan in LDS (wave32: 32 waves on one WGP).
// ---------------------------------------------------------------------------
__device__ __forceinline__ unsigned long long pack_counts(const int* __restrict__ pre,
                                                          const int* __restrict__ seq,
                                                          int b) {
    int pr = pre[b];
    int sq = seq[b];
    unsigned int ext = (unsigned int)(sq - pr);
    unsigned int np  = (unsigned int)(((sq + PAGE_MASK) >> PAGE_SHIFT) -
                                      ((pr + PAGE_MASK) >> PAGE_SHIFT));
    return (unsigned long long)ext | ((unsigned long long)np << 32);
}

__global__ __launch_bounds__(SCAN_THREADS)
void pagealloc_scan_kernel(const int* __restrict__ pre_lens,
                           const int* __restrict__ seq_lens,
                           int* __restrict__ starts,      // [B] exclusive cumsum(ext)
                           int* __restrict__ new_start,   // [B] exclusive cumsum(new_pages)
                           int B) {
    __shared__ unsigned long long lds[SCAN_THREADS];

    const int tid   = threadIdx.x;
    const int items = (B + SCAN_THREADS - 1) / SCAN_THREADS;  // per-thread chunk
    const int base  = tid * items;

    // Local (per-thread) running sum over this thread's contiguous chunk.
    unsigned long long run = 0ull;
    for (int i = 0; i < items; ++i) {
        int b = base + i;
        if (b < B) run += pack_counts(pre_lens, seq_lens, b);
    }
    lds[tid] = run;
    __syncthreads();

    // Hillis-Steele inclusive scan of the 1024 per-thread sums (LDS / ds ops).
    for (int off = 1; off < SCAN_THREADS; off <<= 1) {
        unsigned long long t = (tid >= off) ? lds[tid - off] : 0ull;
        __syncthreads();
        lds[tid] += t;
        __syncthreads();
    }

    // Exclusive prefix for this thread's chunk, then re-walk the chunk.
    unsigned long long acc = (tid > 0) ? lds[tid - 1] : 0ull;
    for (int i = 0; i < items; ++i) {
        int b = base + i;
        if (b < B) {
            starts[b]    = (int)(unsigned int)(acc & 0xffffffffull);
            new_start[b] = (int)(unsigned int)(acc >> 32);
            acc += pack_counts(pre_lens, seq_lens, b);
        }
    }
}

// ---------------------------------------------------------------------------
// Kernel 2: one block per sequence; each block emits that sequence's extend
// token locations contiguously at out[starts[b] .. starts[b]+ext). Writes are
// fully coalesced; 16 consecutive lanes share one free_page entry (L2-hot).
// ---------------------------------------------------------------------------
__global__ __launch_bounds__(256)
void pagealloc_fill_kernel(const int* __restrict__ pre_lens,
                           const int* __restrict__ seq_lens,
                           const int* __restrict__ last_loc,
                           const int* __restrict__ free_page,
                           const int* __restrict__ starts,
                           const int* __restrict__ new_start,
                           int* __restrict__ out,
                           int nfree) {
    const int b  = blockIdx.x;
    const int pr = pre_lens[b];
    const int sq = seq_lens[b];
    const int ext = sq - pr;
    const int st = starts[b];
    const int ns = new_start[b];
    const int ll = last_loc[b];
    const int pages_before = (pr + PAGE_MASK) >> PAGE_SHIFT;
    const int boundary     = pages_before << PAGE_SHIFT;   // ceil(pre/16)*16

    for (int o = (int)threadIdx.x; o < ext; o += (int)blockDim.x) {
        const int p = pr + o;                 // absolute position in sequence
        int v;
        if (p < boundary) {
            // part1: continue filling the existing partially-used page
            v = ll + 1 + o;
        } else {
            // part2/3: token lands on newly allocated page k
            int k   = (p >> PAGE_SHIFT) - pages_before;
            int idx = ns + k;
            idx = idx < 0 ? 0 : (idx >= nfree ? nfree - 1 : idx);   // jnp.clip
            v = (free_page[idx] << PAGE_SHIFT) + (p & PAGE_MASK);
        }
        out[st + o] = v;
    }
}

// ---------------------------------------------------------------------------
// Host-side launch. Inputs (setup_inputs order): pre_lens, seq_lens, last_loc,
// free_page. Output: int32 token locations (written bit-exact via int* view).
// Workspace: 2*B int32 (starts, new_start) — recomputed every call.
// ---------------------------------------------------------------------------
extern "C" void kernel_launch(void* const* d_in, const int* in_sizes, int n_in,
                              void* d_out, int out_size, void* d_ws, size_t ws_size,
                              hipStream_t stream) {
    const int* pre_lens  = (const int*)d_in[0];
    const int* seq_lens  = (const int*)d_in[1];
    const int* last_loc  = (const int*)d_in[2];
    const int* free_page = (const int*)d_in[3];

    const int B     = in_sizes[0];
    const int nfree = in_sizes[3];

    int* starts    = (int*)d_ws;
    int* new_start = starts + B;
    int* out       = (int*)d_out;   // reference output dtype is int32 (4B elements)

    pagealloc_scan_kernel<<<1, SCAN_THREADS, 0, stream>>>(
        pre_lens, seq_lens, starts, new_start, B);

    pagealloc_fill_kernel<<<B, 256, 0, stream>>>(
        pre_lens, seq_lens, last_loc, free_page, starts, new_start, out, nfree);

    (void)n_in; (void)out_size; (void)ws_size;
}